// ChamferDistanceL2_85358180040912
// MI455X (gfx1250) — compile-verified
//
#include <hip/hip_runtime.h>
#include <stdint.h>

typedef float v2f __attribute__((ext_vector_type(2)));
typedef float v8f __attribute__((ext_vector_type(8)));

#define B_      8
#define N_      8192
#define M_      8192
#define MT      512            // xyz2 points staged in LDS per tile
#define NROWS   128            // xyz1 rows owned per block (8 waves * 16)
#define NWAVES  8
#define WSLICE  (MT + 16)      // padded slice stride: halves hit disjoint banks
#define NCHUNK  (MT * 12 / 16) // 16B chunks per raw tile = 384

// Native CDNA5 float-min atomic (no-return) -> global_atomic_min_num_f32.
__device__ __forceinline__ void global_fmin(float* addr, float v) {
  asm volatile("global_atomic_min_num_f32 %0, %1, off"
               :: "v"(addr), "v"(v) : "memory");
}

// CDNA5 async global->LDS copy (tracked by ASYNCcnt, no VGPR data path).
__device__ __forceinline__ void async_load_b128(uint32_t lds_off, const void* gaddr) {
  asm volatile("global_load_async_to_lds_b128 %0, %1, off"
               :: "v"(lds_off), "v"(gaddr) : "memory");
}
__device__ __forceinline__ void wait_async0() {
  asm volatile("s_wait_asynccnt 0" ::: "memory");
}

__global__ __launch_bounds__(256) void chamfer_init(float* __restrict__ dist2) {
  int i = blockIdx.x * 256 + threadIdx.x;
  if (i < B_ * M_) dist2[i] = __builtin_inff();
}

__global__ __launch_bounds__(256) void chamfer_main(const float* __restrict__ xyz1,
                                                    const float* __restrict__ xyz2,
                                                    float* __restrict__ dist1,
                                                    float* __restrict__ dist2) {
  __shared__ __align__(16) float rawy[2][MT * 3];   // 12 KB: async landing pads
  // B fragments pre-formatted per 16-lane half: one LDS load feeds WMMA directly.
  __shared__ __align__(16) float4 fraglo[MT];       //  8 KB: (-2y0, -2y1, |y|^2, 0)
  __shared__ __align__(16) float4 fraghi[MT];       //  8 KB: (-2y2,  1.0, |y|^2, 0)
  __shared__ float wavemin[2 * NWAVES * WSLICE];    // 33 KB: per-half column mins

  const int tid  = threadIdx.x;
  const int wave = tid >> 5;
  const int lane = tid & 31;
  const int col  = lane & 15;
  const bool hi  = lane >= 16;

  const int b      = blockIdx.x / (N_ / NROWS);
  const int nchunk = blockIdx.x % (N_ / NROWS);
  const int nbase  = nchunk * NROWS + wave * 16;

  // --- A fragment: 16x4 f32 (ISA 7.12.2): VGPR0 lanes0-15 K=0 / lanes16-31 K=2,
  //     VGPR1 lanes0-15 K=1 / lanes16-31 K=3.  Row i = [x0, x1, x2, |x|^2].
  const float* xp = xyz1 + ((size_t)b * N_ + (nbase + col)) * 3;
  const float c0 = xp[0], c1 = xp[1], c2 = xp[2];
  const float xnrm = c0 * c0 + c1 * c1 + c2 * c2;
  v2f afrag;
  afrag.x = hi ? c2 : c0;
  afrag.y = hi ? xnrm : c1;

  // Constant per-lane bases: all inner-loop LDS traffic is base + imm offset.
  const float4* fb = (hi ? fraghi : fraglo) + col;
  float* wmptr = &wavemin[(wave * 2 + (hi ? 1 : 0)) * WSLICE + col];

  float rmin[8];
#pragma unroll
  for (int r = 0; r < 8; ++r) rmin[r] = __builtin_inff();

  // One 16-column WMMA step (pure compute; operand already loaded).
  auto step = [&](const float4 t, int j0) {
    v2f bfrag;
    bfrag.x = t.x;                     // K0/K2
    bfrag.y = t.y;                     // K1/K3 (1.0 on hi half)
    v8f acc = {};
    acc = __builtin_amdgcn_wmma_f32_16x16x4_f32(
        false, afrag, false, bfrag, (short)0, acc, false, false);
    float cm = __builtin_inff();
#pragma unroll
    for (int r = 0; r < 8; ++r) {
      const float e = acc[r] + t.z;    // + |y|^2 -> full squared distance
      rmin[r] = fminf(rmin[r], e);
      cm = fminf(cm, e);
    }
    wmptr[j0] = cm;                    // ds_store_b32, imm offset j0*4
  };

  // Async staging of raw xyz2 tiles (16B chunks; tile = 6144B = 384 chunks).
  const char* ybase = (const char*)(xyz2 + (size_t)b * M_ * 3);
  const uint32_t roff[2] = { (uint32_t)(uintptr_t)&rawy[0][0],
                             (uint32_t)(uintptr_t)&rawy[1][0] };
  const int ch0 = tid, ch1 = tid + 256;

#define ISSUE_TILE(mt_, bf_)                                                   \
  do {                                                                         \
    async_load_b128(roff[bf_] + ch0 * 16, ybase + (size_t)(mt_)*12 + ch0 * 16);\
    if (ch1 < NCHUNK)                                                          \
      async_load_b128(roff[bf_] + ch1 * 16, ybase + (size_t)(mt_)*12 + ch1 * 16);\
  } while (0)

  ISSUE_TILE(0, 0);

  int buf = 0;
  for (int mt = 0; mt < M_; mt += MT, buf ^= 1) {
    wait_async0();
    __syncthreads();   // raw[buf] complete for all waves

    // Preprocess raw -> per-half B fragments (once per point per block).
#pragma unroll
    for (int p = tid; p < MT; p += 256) {
      const float y0 = rawy[buf][p * 3 + 0];
      const float y1 = rawy[buf][p * 3 + 1];
      const float y2c = rawy[buf][p * 3 + 2];
      const float ynrm = y0 * y0 + y1 * y1 + y2c * y2c;
      fraglo[p] = make_float4(-2.f * y0, -2.f * y1, ynrm, 0.f);
      fraghi[p] = make_float4(-2.f * y2c, 1.0f, ynrm, 0.f);
    }
    // Overlap next tile's global->LDS copy with this tile's compute.
    if (mt + MT < M_) ISSUE_TILE(mt + MT, buf ^ 1);
    __syncthreads();

    // 32 WMMA steps, software-pipelined: load for step k+1 issues before
    // compute of step k (distinct SSA values -> loads hoist above the wait).
    float4 t = fb[0];
#pragma unroll 4
    for (int j0 = 0; j0 < MT - 16; j0 += 16) {
      const float4 tn = fb[j0 + 16];   // ds_load, imm offset (j0+16)*16
      step(t, j0);
      t = tn;
    }
    step(t, MT - 16);                  // peeled final step (no wrap math)
    __syncthreads();

    // Fold 16 half-slices per column, one native float-min atomic per column.
#pragma unroll
    for (int p = tid; p < MT; p += 256) {
      float v = wavemin[p];
#pragma unroll
      for (int s = 1; s < 2 * NWAVES; ++s) v = fminf(v, wavemin[s * WSLICE + p]);
      global_fmin(dist2 + (size_t)b * M_ + mt + p, v);
    }
    __syncthreads();
  }

  // Row mins: butterfly within each 16-lane half (lanes 0-15 -> M=r,
  // lanes 16-31 -> M=8+r), then exclusive plain stores.
#pragma unroll
  for (int r = 0; r < 8; ++r) {
    float v = rmin[r];
    v = fminf(v, __shfl_xor(v, 1, 32));
    v = fminf(v, __shfl_xor(v, 2, 32));
    v = fminf(v, __shfl_xor(v, 4, 32));
    v = fminf(v, __shfl_xor(v, 8, 32));
    const int row = nbase + (hi ? 8 + r : r);
    if (col == 0)
      dist1[(size_t)b * N_ + row] = v;
  }
}

extern "C" void kernel_launch(void* const* d_in, const int* in_sizes, int n_in,
                              void* d_out, int out_size, void* d_ws, size_t ws_size,
                              hipStream_t stream) {
  (void)in_sizes; (void)n_in; (void)out_size; (void)d_ws; (void)ws_size;
  const float* xyz1 = (const float*)d_in[0];
  const float* xyz2 = (const float*)d_in[1];
  float* dist1 = (float*)d_out;              // [B, N]
  float* dist2 = dist1 + (size_t)B_ * N_;    // [B, M]

  chamfer_init<<<(B_ * M_ + 255) / 256, 256, 0, stream>>>(dist2);
  chamfer_main<<<B_ * (N_ / NROWS), 256, 0, stream>>>(xyz1, xyz2, dist1, dist2);
}